// DTMLayer_10325101379889
// MI455X (gfx1250) — compile-verified
//
#include <hip/hip_runtime.h>
#include <math.h>

// DTM layer: B=2, N=4096 (64x64 grid), M0=0.3.
// d2[i,j] = |g_i|^2 + |x_j|^2 - 2 g_i.x_j  via V_WMMA_F32_16X16X4_F32 (K=2 padded to 4).
// Selection without sorting: bisection on threshold t (tie-invariant DTM formula),
// then exact snap of t* and strict-below weighted sums.

typedef __attribute__((ext_vector_type(2))) float v2f;
typedef __attribute__((ext_vector_type(8))) float v8f;

#define NPTS 4096
#define ROWS 4100          // LDS row stride (floats); 8*ROWS % 64 == 32 -> halves hit disjoint banks
#define M0C  0.3f

struct Smem {
    float px[NPTS];
    float py[NPTS];
    float w[NPTS];
    float gn[16];          // |g_m|^2 for the 16 queries of this tile
    float wbsum;           // total weight (atomic LDS accumulation)
    float d2[16 * ROWS];   // distance-squared rows for 16 queries
};

// One 16x16 distance tile: WMMA cross term + |g|^2 + |x|^2 assembly, store to LDS.
// bmask zeroes the B fragment in lanes 16-31 (K=2,3 zero padding) without EXEC divergence.
__device__ __forceinline__ void dist_tile(Smem* s, const v2f& a, float bmask,
                                          const float* __restrict__ gnv,
                                          int dbase, int nn, int j0)
{
    const int j = j0 + nn;
    const float pxv = s->px[j];
    const float pyv = s->py[j];
    v2f bf;
    bf[0] = pxv * bmask;
    bf[1] = pyv * bmask;
    v8f c = {};
    c = __builtin_amdgcn_wmma_f32_16x16x4_f32(
            /*neg_a=*/false, a, /*neg_b=*/false, bf,
            /*c_mod=*/(short)0, c, /*reuse_a=*/false, /*reuse_b=*/false);
    const float xn2 = pxv * pxv + pyv * pyv;
    #pragma unroll
    for (int r = 0; r < 8; ++r) {
        const float d2v = fmaxf(gnv[r] + xn2 - 2.0f * c[r], 0.0f);
        s->d2[dbase + r * ROWS + j0] = d2v;
    }
}

__global__ void __launch_bounds__(256) dtm_kernel(const float* __restrict__ inp,
                                                  const float* __restrict__ weight,
                                                  const float* __restrict__ grid,
                                                  float* __restrict__ out)
{
    extern __shared__ char smem_raw[];
    Smem* s = (Smem*)smem_raw;

    const int tid  = threadIdx.x;
    const int lane = tid & 31;          // wave32
    const int wv   = tid >> 5;          // 8 waves
    const int b    = blockIdx.x >> 8;   // 256 query tiles per batch
    const int q0   = (blockIdx.x & 255) << 4;

    const float* xb = inp    + (size_t)b * NPTS * 2;
    const float* wp = weight + (size_t)b * NPTS;

    // ---- Phase 1: stage batch points + weights into LDS (coalesced float2) ----
    for (int j = tid; j < NPTS; j += 256) {
        float2 p = ((const float2*)xb)[j];
        s->px[j] = p.x;
        s->py[j] = p.y;
        s->w[j]  = wp[j];
    }

    // A fragment: 16x4 f32. Lanes 0-15: VGPR0=K0, VGPR1=K1 (the 2 coords);
    // lanes 16-31: K2,K3 = zero padding.
    const bool  lowHalf = (lane < 16);
    const float bmask   = lowHalf ? 1.0f : 0.0f;
    v2f a; a[0] = 0.0f; a[1] = 0.0f;
    if (lowHalf) {
        float2 g = ((const float2*)grid)[q0 + lane];
        a[0] = g.x; a[1] = g.y;
        if (wv == 0) s->gn[lane] = g.x * g.x + g.y * g.y;
    }
    if (tid == 0) s->wbsum = 0.0f;
    __syncthreads();

    // ---- Phase 2a: total-weight reduction (wave butterfly + ds_add_f32) ----
    float part = 0.0f;
    for (int j = tid; j < NPTS; j += 256) part += s->w[j];
    #pragma unroll
    for (int off = 16; off >= 1; off >>= 1) part += __shfl_xor(part, off, 32);
    if (lane == 0) atomicAdd(&s->wbsum, part);

    // ---- Phase 2b: WMMA distance tiles (each wave: 32 of 256 j-tiles, 2 per iter) ----
    const int nn    = lane & 15;
    const int mbase = lowHalf ? 0 : 8;   // D layout: VGPR r -> M=r (lanes 0-15) / M=8+r
    const int dbase = mbase * ROWS + nn;
    float gnv[8];                         // loop-invariant |g|^2 broadcasts, hoisted
    #pragma unroll
    for (int r = 0; r < 8; ++r) gnv[r] = s->gn[mbase + r];

    for (int jt = wv * 2; jt < NPTS / 16; jt += 16) {
        dist_tile(s, a, bmask, gnv, dbase, nn, (jt    ) << 4);
        dist_tile(s, a, bmask, gnv, dbase, nn, (jt + 1) << 4);
    }
    __syncthreads();

    // ---- Phase 3: per-query weighted-quantile selection (2 queries per wave) ----
    const float wbv = M0C * s->wbsum;
    for (int m = wv; m < 16; m += 8) {
        const float* row = &s->d2[m * ROWS];

        // Bisection on t in d^2 space: invariant W(lo) < wbv <= W(hi).
        float lo = 0.0f, hi = 64.0f;
        for (int it = 0; it < 40; ++it) {
            const float t = 0.5f * (lo + hi);
            float acc = 0.0f;
            for (int j = lane; j < NPTS; j += 32)
                acc += (row[j] <= t) ? s->w[j] : 0.0f;
            #pragma unroll
            for (int off = 16; off >= 1; off >>= 1) acc += __shfl_xor(acc, off, 32);
            if (acc >= wbv) hi = t; else lo = t;
        }

        // Snap to the exact crossing distance: smallest d^2 strictly above lo.
        float tstar = 3.4e38f;
        for (int j = lane; j < NPTS; j += 32) {
            const float d = row[j];
            tstar = (d > lo) ? fminf(tstar, d) : tstar;
        }
        #pragma unroll
        for (int off = 16; off >= 1; off >>= 1)
            tstar = fminf(tstar, __shfl_xor(tstar, off, 32));

        // Strict-below sums; DTM = S< + t*^2 (wbv - W<)   (tie-invariant form).
        float Wlt = 0.0f, Slt = 0.0f;
        for (int j = lane; j < NPTS; j += 32) {
            const float d = row[j];
            if (d < tstar) { Wlt += s->w[j]; Slt += s->w[j] * d; }
        }
        #pragma unroll
        for (int off = 16; off >= 1; off >>= 1) {
            Wlt += __shfl_xor(Wlt, off, 32);
            Slt += __shfl_xor(Slt, off, 32);
        }

        if (lane == 0) {
            const float dtm = Slt + tstar * (wbv - Wlt);
            out[(size_t)b * NPTS + q0 + m] = sqrtf(fmaxf(dtm, 0.0f) / wbv);
        }
    }
}

extern "C" void kernel_launch(void* const* d_in, const int* in_sizes, int n_in,
                              void* d_out, int out_size, void* d_ws, size_t ws_size,
                              hipStream_t stream)
{
    const float* inp    = (const float*)d_in[0]; // (B, N, 2)
    const float* weight = (const float*)d_in[1]; // (B, N)
    const float* grid   = (const float*)d_in[2]; // (N, 2)
    float* out = (float*)d_out;                  // (B, N)

    const int B = in_sizes[1] / NPTS;            // = 2
    const size_t smem = sizeof(Smem);            // ~305 KB, fits 320 KB/WGP
    dim3 g(B * (NPTS / 16));                     // 512 workgroups
    dtm_kernel<<<g, 256, smem, stream>>>(inp, weight, grid, out);
}